// JPEGProxy_77051713290823
// MI455X (gfx1250) — compile-verified
//
#include <hip/hip_runtime.h>

// JPEG proxy, fully fused: RGB->YUV -> blockwise 8x8 DCT -> quant(round-STE)/dequant
// -> IDCT -> YUV->RGB -> clip. 192MB HBM traffic => ~8.2us floor at 23.3TB/s.
// DCT/IDCT via V_WMMA_F32_16X16X4_F32 (fp32), 16x16 tile = 4 independent 8x8
// blocks using block-diagonal D = diag(C8,C8). Every stage: X = M * (plane),
// stored transposed (contiguous b128) so the next stage reads natural columns.
// Each block: 3 wave32 (one per channel) x 4 tiles (64x16 strip), setup amortized.

typedef __attribute__((ext_vector_type(2))) float v2f;
typedef __attribute__((ext_vector_type(8))) float v8f;

__constant__ float C8[8][8] = {
  { 0.353553391f, 0.353553391f, 0.353553391f, 0.353553391f, 0.353553391f, 0.353553391f, 0.353553391f, 0.353553391f},
  { 0.490392640f, 0.415734806f, 0.277785117f, 0.097545161f,-0.097545161f,-0.277785117f,-0.415734806f,-0.490392640f},
  { 0.461939766f, 0.191341716f,-0.191341716f,-0.461939766f,-0.461939766f,-0.191341716f, 0.191341716f, 0.461939766f},
  { 0.415734806f,-0.097545161f,-0.490392640f,-0.277785117f, 0.277785117f, 0.490392640f, 0.097545161f,-0.415734806f},
  { 0.353553391f,-0.353553391f,-0.353553391f, 0.353553391f, 0.353553391f,-0.353553391f,-0.353553391f, 0.353553391f},
  { 0.277785117f,-0.490392640f, 0.097545161f, 0.415734806f,-0.415734806f,-0.097545161f, 0.490392640f,-0.277785117f},
  { 0.191341716f,-0.461939766f, 0.461939766f,-0.191341716f,-0.191341716f, 0.461939766f,-0.461939766f, 0.191341716f},
  { 0.097545161f,-0.277785117f, 0.415734806f,-0.490392640f, 0.490392640f,-0.415734806f, 0.277785117f,-0.097545161f},
};

__constant__ float IJG[8][8] = {
  {16,11,10,16,24,40,51,61},{12,12,14,19,26,58,60,55},
  {14,13,16,24,40,57,69,56},{14,17,22,29,51,87,80,62},
  {18,22,37,56,68,109,103,77},{24,35,55,64,81,104,113,92},
  {49,64,78,87,103,121,120,101},{72,92,95,98,112,100,103,99},
};

__device__ __forceinline__ float Dm(int i, int j) {
  return ((i ^ j) & 8) ? 0.0f : C8[i & 7][j & 7];
}

#define LSTRIDE 20   // padded LDS row stride (floats): 16B rows, conflict-free
#define TILES   4    // 16x16 tiles per block (64x16 strip)

// 16x16x16 fp32 matmul: 4 chained V_WMMA_F32_16X16X4_F32
__device__ __forceinline__ v8f wmma16(const v2f* A, const v2f* B) {
  v8f acc = (v8f){};
#pragma unroll
  for (int kc = 0; kc < 4; ++kc)
    acc = __builtin_amdgcn_wmma_f32_16x16x4_f32(false, A[kc], false, B[kc],
                                                (short)0, acc, false, false);
  return acc;
}

// B operand, natural column read: B[k][n] = P[k*LSTRIDE + n], k = 4kc+2h+j
__device__ __forceinline__ void load_cols(const float* __restrict__ P,
                                          int h, int ln, v2f* bv) {
#pragma unroll
  for (int kc = 0; kc < 4; ++kc) {
    int k0 = 4 * kc + 2 * h;
    bv[kc].x = P[(k0 + 0) * LSTRIDE + ln];
    bv[kc].y = P[(k0 + 1) * LSTRIDE + ln];
  }
}

// Store C-layout acc transposed: plane[N][M] = X[M][N]  (two b128 per lane)
__device__ __forceinline__ void store_T(float* __restrict__ P, int h, int ln,
                                        v8f acc) {
  float4 lo = make_float4(acc[0], acc[1], acc[2], acc[3]);
  float4 hi = make_float4(acc[4], acc[5], acc[6], acc[7]);
  *(float4*)&P[ln * LSTRIDE + 8 * h]     = lo;
  *(float4*)&P[ln * LSTRIDE + 8 * h + 4] = hi;
}

__global__ __launch_bounds__(96) void jpeg_proxy_kernel(
    const float* __restrict__ x, const int* __restrict__ qf_p,
    float* __restrict__ out) {
  __shared__ float planes[3][16 * LSTRIDE];

  const int tid  = threadIdx.x;
  const int wave = tid >> 5;            // channel owned by this wave
  const int lane = tid & 31;
  const int h    = lane >> 4;           // WMMA lane-half
  const int ln   = lane & 15;

  // grid: 8 batches x 64 rows-of-tiles x 16 strips
  const int bidx = blockIdx.x;
  const int b    = bidx >> 10;
  const int rest = bidx & 1023;
  const int y0   = (rest >> 4) << 4;
  const int sx   = (rest & 15) << 6;    // strip x start (64-wide)

  const size_t chS = 1024ull * 1024ull;
  const float* src = x   + ((size_t)b * 3 + wave) * chS + (size_t)y0 * 1024 + sx;
  float*       dst = out + ((size_t)b * 3 + wave) * chS + (size_t)y0 * 1024 + sx;
  float* P = planes[wave];

  // ---- one-time setup (amortized over 4 tiles) ----
  // float4-group mapping: group i in {0,1}: row = fy0 + 8i, col4 = fx
  const int fy0 = lane >> 2;
  const int fx  = (lane & 3) << 2;
  const int g0  = fy0 * 1024 + fx,        g1 = g0 + 8 * 1024;
  const int l0  = fy0 * LSTRIDE + fx,     l1 = l0 + 8 * LSTRIDE;

  // A-fragments for D and D^T (16x4 fp32 A layout: vgpr j = Mat[m][4kc+2h+j])
  v2f aD[4], aDT[4];
#pragma unroll
  for (int kc = 0; kc < 4; ++kc) {
    int k0 = 4 * kc + 2 * h;
    aD[kc].x  = Dm(ln, k0);  aD[kc].y  = Dm(ln, k0 + 1);
    aDT[kc].x = Dm(k0, ln);  aDT[kc].y = Dm(k0 + 1, ln);
  }

  // quant row: element (M=r+8h, N=ln) of coeff^T == coeff[p=ln][q=r+8h]
  //   -> qt = qtable[ln&7][r]
  const int   qf    = *qf_p;
  const float scale = (qf < 50) ? (5000.0f / (float)qf) : (200.0f - 2.0f * (float)qf);
  float qv[8], rqv[8];
#pragma unroll
  for (int r = 0; r < 8; ++r) {
    float q = (scale * IJG[ln & 7][r] + 50.0f) * 0.01f;
    float rc = __builtin_amdgcn_rcpf(q);
    rc = rc * (2.0f - q * rc);          // 1 Newton step: ~full fp32 precision
    qv[r] = q;  rqv[r] = rc;
  }

  const float w0  = (wave == 0) ? 0.299f : (wave == 1) ? -0.168736f :  0.5f;
  const float w1  = (wave == 0) ? 0.587f : (wave == 1) ? -0.331264f : -0.418688f;
  const float w2  = (wave == 0) ? 0.114f : (wave == 1) ?  0.5f      : -0.081312f;
  const float ofc = (wave == 0) ? 0.0f : 128.0f;
  const float mu  = (wave == 0) ? 0.0f   : (wave == 1) ? -0.344136f : 1.772f;
  const float mv  = (wave == 0) ? 1.402f : (wave == 1) ? -0.714136f : 0.0f;

  // prefetch tile 0
  float4 r0 = *(const float4*)(src + g0);
  float4 r1 = *(const float4*)(src + g1);

  for (int t = 0; t < TILES; ++t) {
    // ---- phase 0: registers -> LDS (raw channel plane) ----
    *(float4*)&P[l0] = r0;
    *(float4*)&P[l1] = r1;
    if (t + 1 < TILES) {                 // prefetch next tile during compute
      r0 = *(const float4*)(src + (t + 1) * 16 + g0);
      r1 = *(const float4*)(src + (t + 1) * 16 + g1);
    }
    __syncthreads();

    // ---- phase 1: RGB->YUV computed directly into stage-1 B fragments ----
    // lane (h,ln) feeds B[k][ln] for k = 4kc+2h+j  (8 column elements)
    v2f bv[4];
#pragma unroll
    for (int kc = 0; kc < 4; ++kc) {
#pragma unroll
      for (int j = 0; j < 2; ++j) {
        int o = (4 * kc + 2 * h + j) * LSTRIDE + ln;
        float val = w0 * planes[0][o] + w1 * planes[1][o] + w2 * planes[2][o] + ofc;
        if (j == 0) bv[kc].x = val; else bv[kc].y = val;
      }
    }
    __syncthreads();                     // all raw-RGB reads done before overwrite

    // ---- stage 1: X1 = D*Y            -> plane = X1^T ----
    v8f acc = wmma16(aD, bv);
    store_T(P, h, ln, acc);

    // ---- stage 2: X2 = D*X1^T = coeff^T ; quantize ; -> plane = Q ----
    load_cols(P, h, ln, bv);
    acc = wmma16(aD, bv);
#pragma unroll
    for (int r = 0; r < 8; ++r)
      acc[r] = rintf(acc[r] * rqv[r]) * qv[r];   // RNE matches jnp.round
    store_T(P, h, ln, acc);

    // ---- stage 3: X3 = D^T*Q          -> plane = X3^T ----
    load_cols(P, h, ln, bv);
    acc = wmma16(aDT, bv);
    store_T(P, h, ln, acc);

    // ---- stage 4: X4 = D^T*X3^T = R^T -> plane = R (natural) ----
    load_cols(P, h, ln, bv);
    acc = wmma16(aDT, bv);
    store_T(P, h, ln, acc);
    __syncthreads();

    // ---- phase 5: YUV->RGB, clip, coalesced b128 store ----
#pragma unroll
    for (int i = 0; i < 2; ++i) {
      int lo = i ? l1 : l0;
      float4 yv = *(const float4*)&planes[0][lo];
      float4 uv = *(const float4*)&planes[1][lo];
      float4 vv = *(const float4*)&planes[2][lo];
      float ov[4];
      const float* yp = &yv.x; const float* up = &uv.x; const float* vp = &vv.x;
#pragma unroll
      for (int j = 0; j < 4; ++j) {
        float val = yp[j] + mu * (up[j] - 128.0f) + mv * (vp[j] - 128.0f);
        ov[j] = fminf(fmaxf(val, 0.0f), 255.0f);
      }
      float4 o4 = make_float4(ov[0], ov[1], ov[2], ov[3]);
      *(float4*)(dst + t * 16 + (i ? g1 : g0)) = o4;
    }
    __syncthreads();                     // protect planes before next phase 0
  }
}

extern "C" void kernel_launch(void* const* d_in, const int* in_sizes, int n_in,
                              void* d_out, int out_size, void* d_ws, size_t ws_size,
                              hipStream_t stream) {
  const float* x  = (const float*)d_in[0];
  const int*   qf = (const int*)d_in[1];
  float*       out = (float*)d_out;
  (void)in_sizes; (void)n_in; (void)out_size; (void)d_ws; (void)ws_size;
  // 8 batches x 64 tile-rows x 16 strips; 96 threads = 3 wave32
  dim3 grid(8 * 64 * 16);
  jpeg_proxy_kernel<<<grid, 96, 0, stream>>>(x, qf, out);
}